// NormalShader_5282809774751
// MI455X (gfx1250) — compile-verified
//
#include <hip/hip_runtime.h>

typedef __attribute__((ext_vector_type(4))) float v4f;

#define TPB 256
#define KF  8

#define SIGMA_INV 1.0e4f          // 1/1e-4
#define GAMMA_INV 1.0e4f          // 1/1e-4
#define ZSPAN_INV (1.0f / 99.0f)  // 1/(ZFAR-ZNEAR)
#define ZFAR_C    100.0f
#define EPS_C     1e-10f

// CDNA5 async DMA: global memory -> LDS, 16 bytes/lane, tracked by ASYNCcnt.
// TH_LOAD_NT: streaming read-once data must not evict the L2-resident
// faces/vertex_normals gather tables (1.8 MB reused ~8M times).
__device__ __forceinline__ void async_g2l_b128(unsigned lds_byte_addr,
                                               unsigned long long gaddr) {
  asm volatile("global_load_async_to_lds_b128 %0, %1, off th:TH_LOAD_NT"
               :: "v"(lds_byte_addr), "v"(gaddr) : "memory");
}

__device__ __forceinline__ void wait_asynccnt0() {
  asm volatile("s_wait_asynccnt 0" ::: "memory");
}

__device__ __forceinline__ float fast_rcp(float x) {
  return __builtin_amdgcn_rcpf(x);   // v_rcp_f32, ~1 ulp
}

struct alignas(4) F3 { float x, y, z; };
struct alignas(4) I3 { int   x, y, z; };

__global__ __launch_bounds__(TPB)
void normal_shader_kernel(const int*   __restrict__ p2f,
                          const float* __restrict__ bary,
                          const float* __restrict__ dists,
                          const float* __restrict__ zbuf,
                          const int*   __restrict__ faces,
                          const float* __restrict__ vn,
                          float*       __restrict__ out,
                          int npix)
{
  __shared__ float s_bary[TPB * KF * 3]; // 24 KB
  __shared__ int   s_p2f [TPB * KF];     //  8 KB
  __shared__ float s_dist[TPB * KF];     //  8 KB
  __shared__ float s_zbuf[TPB * KF];     //  8 KB   -> 48 KB total

  const int tid = threadIdx.x;
  const int p0  = blockIdx.x * TPB;
  const int pix = p0 + tid;

  // ---------- Stage this block's streaming inputs into LDS (async DMA) ------
  {
    // bary: 96 B/pixel -> 6 b128 units per thread
    unsigned long long base = (unsigned long long)bary;
    unsigned long long lim  = (unsigned long long)npix * 96ull - 16ull;
    unsigned lbase = (unsigned)(unsigned long long)(void*)s_bary;
#pragma unroll
    for (int i = 0; i < 6; ++i) {
      unsigned u = (unsigned)tid + (unsigned)i * TPB;
      unsigned long long go = (unsigned long long)p0 * 96ull +
                              (unsigned long long)u * 16ull;
      if (go > lim) go = lim;
      async_g2l_b128(lbase + u * 16u, base + go);
    }
    // p2f: 32 B/pixel -> 2 units per thread
    base  = (unsigned long long)p2f;
    lim   = (unsigned long long)npix * 32ull - 16ull;
    lbase = (unsigned)(unsigned long long)(void*)s_p2f;
#pragma unroll
    for (int i = 0; i < 2; ++i) {
      unsigned u = (unsigned)tid + (unsigned)i * TPB;
      unsigned long long go = (unsigned long long)p0 * 32ull +
                              (unsigned long long)u * 16ull;
      if (go > lim) go = lim;
      async_g2l_b128(lbase + u * 16u, base + go);
    }
    // dists: 32 B/pixel
    base  = (unsigned long long)dists;
    lim   = (unsigned long long)npix * 32ull - 16ull;
    lbase = (unsigned)(unsigned long long)(void*)s_dist;
#pragma unroll
    for (int i = 0; i < 2; ++i) {
      unsigned u = (unsigned)tid + (unsigned)i * TPB;
      unsigned long long go = (unsigned long long)p0 * 32ull +
                              (unsigned long long)u * 16ull;
      if (go > lim) go = lim;
      async_g2l_b128(lbase + u * 16u, base + go);
    }
    // zbuf: 32 B/pixel
    base  = (unsigned long long)zbuf;
    lim   = (unsigned long long)npix * 32ull - 16ull;
    lbase = (unsigned)(unsigned long long)(void*)s_zbuf;
#pragma unroll
    for (int i = 0; i < 2; ++i) {
      unsigned u = (unsigned)tid + (unsigned)i * TPB;
      unsigned long long go = (unsigned long long)p0 * 32ull +
                              (unsigned long long)u * 16ull;
      if (go > lim) go = lim;
      async_g2l_b128(lbase + u * 16u, base + go);
    }
  }

  wait_asynccnt0();
  __syncthreads();

  if (pix >= npix) return;

  // ---------- Pass 1: gather + interpolate normals, prob, z_inv, running max
  float pnx[KF], pny[KF], pnz[KF], prob[KF], zi[KF];
  float keep = 1.0f;   // prod(1 - prob)
  float zmax = 0.0f;

#pragma unroll
  for (int k = 0; k < KF; ++k) {
    const int s = tid * KF + k;
    const int f = s_p2f[s];
    const bool m = (f >= 0);
    const int idx = m ? f : 0;

    // faces[idx] : 3 vertex indices (global_load_b96, L2-resident)
    const I3 fv = ((const I3*)faces)[idx];

    const float b0 = s_bary[s * 3 + 0];
    const float b1 = s_bary[s * 3 + 1];
    const float b2 = s_bary[s * 3 + 2];

    // vertex_normals gathers (global_load_b96, L2-resident)
    const F3 n0 = ((const F3*)vn)[fv.x];
    const F3 n1 = ((const F3*)vn)[fv.y];
    const F3 n2 = ((const F3*)vn)[fv.z];

    float nx = b0 * n0.x + b1 * n1.x + b2 * n2.x;
    float ny = b0 * n0.y + b1 * n1.y + b2 * n2.y;
    float nz = b0 * n0.z + b1 * n1.z + b2 * n2.z;
    if (!m) { nx = 0.0f; ny = 0.0f; nz = 0.0f; }

    // prob = sigmoid(-d/sigma)*mask = mask / (1 + exp(d/sigma))
    const float d  = s_dist[s];
    const float pr = m ? fast_rcp(1.0f + __expf(d * SIGMA_INV)) : 0.0f;
    keep *= (1.0f - pr);

    const float z  = s_zbuf[s];
    const float zv = m ? (ZFAR_C - z) * ZSPAN_INV : 0.0f;
    zmax = fmaxf(zmax, zv);

    pnx[k] = nx; pny[k] = ny; pnz[k] = nz;
    prob[k] = pr; zi[k] = zv;
  }

  zmax = fmaxf(zmax, EPS_C);
  const float delta = fmaxf(__expf((EPS_C - zmax) * GAMMA_INV), EPS_C);

  // ---------- Pass 2: softmax blend (bg color = 1,1,1) ----------------------
  float denom = delta;
  float ax = delta, ay = delta, az = delta; // delta * bg
#pragma unroll
  for (int k = 0; k < KF; ++k) {
    const float w = prob[k] * __expf((zi[k] - zmax) * GAMMA_INV);
    denom += w;
    ax += w * pnx[k];
    ay += w * pny[k];
    az += w * pnz[k];
  }

  const float inv = fast_rcp(denom);
  v4f o;
  o.x = ax * inv;
  o.y = ay * inv;
  o.z = az * inv;
  o.w = 1.0f - keep;
  // streaming NT store: 16 MB result must not pollute near caches
  __builtin_nontemporal_store(o, (v4f*)out + pix);
}

extern "C" void kernel_launch(void* const* d_in, const int* in_sizes, int n_in,
                              void* d_out, int out_size, void* d_ws, size_t ws_size,
                              hipStream_t stream) {
  const int*   p2f   = (const int*)  d_in[0];
  const float* bary  = (const float*)d_in[1];
  const float* dists = (const float*)d_in[2];
  const float* zbuf  = (const float*)d_in[3];
  const int*   faces = (const int*)  d_in[4];
  const float* vn    = (const float*)d_in[5];
  float*       out   = (float*)d_out;

  const int npix   = in_sizes[0] / KF;               // N*H*W
  const int blocks = (npix + TPB - 1) / TPB;

  hipLaunchKernelGGL(normal_shader_kernel, dim3(blocks), dim3(TPB), 0, stream,
                     p2f, bary, dists, zbuf, faces, vn, out, npix);
}